// GCNLayer_77403900608994
// MI455X (gfx1250) — compile-verified
//
#include <hip/hip_runtime.h>
#include <hip/hip_bf16.h>

#define F 128
#define BN_EPS 1e-5f

typedef __attribute__((ext_vector_type(2))) float v2f;
typedef __attribute__((ext_vector_type(8))) float v8f;

__device__ __forceinline__ void atomAddF(float* p, float v) {
    __hip_atomic_fetch_add(p, v, __ATOMIC_RELAXED, __HIP_MEMORY_SCOPE_AGENT);
}

// ---------------- degree / dinv ----------------

__global__ void k_deg_init(float* __restrict__ deg, int n) {
    int i = blockIdx.x * blockDim.x + threadIdx.x;
    if (i < n) deg[i] = 1.0f;  // self-loop contributes 1
}

__global__ void k_deg_edges(const int* __restrict__ ei, float* __restrict__ deg, int E) {
    int e = blockIdx.x * blockDim.x + threadIdx.x;
    if (e < E) atomAddF(&deg[ei[E + e]], 1.0f);  // dst row of edge_index
}

__global__ void k_dinv(float* __restrict__ deg, int n) {
    int i = blockIdx.x * blockDim.x + threadIdx.x;
    if (i < n) deg[i] = rsqrtf(deg[i]);  // deg >= 1 always (self-loop)
}

__global__ void k_zero_stats(float* __restrict__ s) {
    s[threadIdx.x] = 0.0f;  // 256 floats: gsum[128] ++ gsumsq[128]
}

// ---------------- GEMM: x_lin = X @ W via V_WMMA_F32_16X16X4_F32 ----------------
// One wave computes a 16-row x 128-col strip: 8 accumulator tiles of 16x16,
// K swept in steps of 4. X read exactly once; W (64KB) is cache-resident.

__global__ __launch_bounds__(256) void k_gemm(const float* __restrict__ X,
                                              const float* __restrict__ W,
                                              float* __restrict__ Y, int nrows) {
    int wid  = (blockIdx.x * blockDim.x + threadIdx.x) >> 5;  // global wave id
    int lane = threadIdx.x & 31;
    int m0 = wid * 16;
    if (m0 >= nrows) return;  // wave-uniform: EXEC stays all-ones for WMMA

    int mrow = lane & 15;            // M (for A) / N (for B) index
    int kb   = (lane >> 4) << 1;     // 0 (lanes 0-15) or 2 (lanes 16-31)

    const float* xrow = X + (size_t)(m0 + mrow) * F;

    v8f acc[8];
#pragma unroll
    for (int n = 0; n < 8; ++n) acc[n] = (v8f){0.f,0.f,0.f,0.f,0.f,0.f,0.f,0.f};

    for (int k = 0; k < F; k += 4) {
        v2f a;
        a.x = xrow[k + kb];
        a.y = xrow[k + kb + 1];
        const float* wr0 = W + (size_t)(k + kb) * F;
        const float* wr1 = wr0 + F;
#pragma unroll
        for (int n = 0; n < 8; ++n) {
            v2f b;
            b.x = wr0[n * 16 + mrow];
            b.y = wr1[n * 16 + mrow];
            acc[n] = __builtin_amdgcn_wmma_f32_16x16x4_f32(
                false, a, false, b, (short)0, acc[n], false, false);
        }
    }

    // D layout: VGPR v, lanes 0-15 -> M=v, N=lane; lanes 16-31 -> M=v+8, N=lane-16
    int mhi = (lane >> 4) << 3;  // 0 or 8
#pragma unroll
    for (int v = 0; v < 8; ++v) {
        float* yr = Y + (size_t)(m0 + mhi + v) * F + mrow;
#pragma unroll
        for (int n = 0; n < 8; ++n) yr[n * 16] = acc[n][v];
    }
}

// ---------------- agg init with self-loop term: agg = x_lin * dinv^2 ----------------

__global__ void k_self_init(const float* __restrict__ xlin, const float* __restrict__ dinv,
                            float* __restrict__ agg, int n) {
    int t = blockIdx.x * blockDim.x + threadIdx.x;  // n*32 threads, float4 each
    if (t >= n * 32) return;
    int node = t >> 5;
    float s = dinv[node];
    s = s * s;
    const float4 v = *(const float4*)(xlin + (size_t)t * 4);
    float4 r = make_float4(v.x * s, v.y * s, v.z * s, v.w * s);
    *(float4*)(agg + (size_t)t * 4) = r;
}

// ---------------- edge scatter: one wave per edge ----------------

__global__ __launch_bounds__(256) void k_scatter(const int* __restrict__ ei,
                                                 const float* __restrict__ dinv,
                                                 const float* __restrict__ xlin,
                                                 float* __restrict__ agg, int E) {
    int t = blockIdx.x * blockDim.x + threadIdx.x;
    int e = t >> 5;        // all 32 lanes of a wave share one edge
    int lane = t & 31;
    if (e >= E) return;
    int s = ei[e];
    int d = ei[E + e];
    float nrm = dinv[s] * dinv[d];
    const float4 v = *(const float4*)(xlin + (size_t)s * F + lane * 4);
    float* ap = agg + (size_t)d * F + lane * 4;
    atomAddF(ap + 0, v.x * nrm);
    atomAddF(ap + 1, v.y * nrm);
    atomAddF(ap + 2, v.z * nrm);
    atomAddF(ap + 3, v.w * nrm);
}

// ---------------- BN column stats: sum and sum-of-squares per feature ----------------

__global__ __launch_bounds__(256) void k_stats(const float* __restrict__ agg,
                                               float* __restrict__ gsum,
                                               float* __restrict__ gsumsq, int n) {
    int f   = threadIdx.x & 127;                       // feature (coalesced)
    int row = blockIdx.x * 2 + (threadIdx.x >> 7);     // row stream
    int stride = gridDim.x * 2;
    float s = 0.f, sq = 0.f;
    for (int r = row; r < n; r += stride) {
        float v = agg[(size_t)r * F + f];
        s += v;
        sq += v * v;
    }
    atomAddF(&gsum[f], s);
    atomAddF(&gsumsq[f], sq);
}

// ---------------- fused BN + ReLU + residual (bias b cancels inside BN) ----------------

__global__ void k_final(const float* __restrict__ agg, const float* __restrict__ x,
                        const float* __restrict__ gsum, const float* __restrict__ gsumsq,
                        const float* __restrict__ gamma, const float* __restrict__ beta,
                        float* __restrict__ out, int n) {
    int idx = blockIdx.x * blockDim.x + threadIdx.x;
    if (idx >= n * F) return;
    int f = idx & 127;
    float invn = 1.0f / (float)n;
    float mean = gsum[f] * invn;
    float var  = gsumsq[f] * invn - mean * mean;  // biased variance (jnp.var default)
    float inv  = rsqrtf(var + BN_EPS);
    float h = (agg[idx] - mean) * inv * gamma[f] + beta[f];
    out[idx] = fmaxf(h, 0.0f) + x[idx];
}

// ---------------- launcher ----------------

extern "C" void kernel_launch(void* const* d_in, const int* in_sizes, int n_in,
                              void* d_out, int out_size, void* d_ws, size_t ws_size,
                              hipStream_t stream) {
    const float* x     = (const float*)d_in[0];
    const int*   ei    = (const int*)d_in[1];
    const float* W     = (const float*)d_in[2];
    // d_in[3] = bias b: mathematically cancels in BatchNorm, skipped
    const float* gamma = (const float*)d_in[4];
    const float* beta  = (const float*)d_in[5];
    float* out = (float*)d_out;

    const int N = in_sizes[0] / F;
    const int E = in_sizes[1] / 2;

    float* ws     = (float*)d_ws;
    float* xlin   = ws;                          // N*F
    float* agg    = xlin + (size_t)N * F;        // N*F
    float* dinv   = agg + (size_t)N * F;         // N
    float* gsum   = dinv + N;                    // 128
    float* gsumsq = gsum + F;                    // 128

    const int B = 256;

    k_deg_init<<<(N + B - 1) / B, B, 0, stream>>>(dinv, N);
    k_zero_stats<<<1, 256, 0, stream>>>(gsum);
    k_deg_edges<<<(E + B - 1) / B, B, 0, stream>>>(ei, dinv, E);
    k_dinv<<<(N + B - 1) / B, B, 0, stream>>>(dinv, N);

    int waves = (N + 15) / 16;                     // one wave per 16-row strip
    k_gemm<<<(waves + 7) / 8, B, 0, stream>>>(x, W, xlin, N);

    k_self_init<<<((size_t)N * 32 + B - 1) / B, B, 0, stream>>>(xlin, dinv, agg, N);
    k_scatter<<<((size_t)E * 32 + B - 1) / B, B, 0, stream>>>(ei, dinv, xlin, agg, E);
    k_stats<<<512, B, 0, stream>>>(agg, gsum, gsumsq, N);
    k_final<<<((size_t)N * F + B - 1) / B, B, 0, stream>>>(agg, x, gsum, gsumsq,
                                                           gamma, beta, out, N);
}